// GCNVariant_31610959298973
// MI455X (gfx1250) — compile-verified
//
#include <hip/hip_runtime.h>
#include <hip/hip_bf16.h>
#include <math.h>

#define FIN   128
#define FHID  128
#define FOUT  32
#define BN_EPS 1e-5f

typedef float v2f __attribute__((ext_vector_type(2)));
typedef float v8f __attribute__((ext_vector_type(8)));

__device__ __forceinline__ void atomAddF(float* p, float v) {
  // relaxed, device-scope -> hardware global_atomic_add_f32 (no return)
  __hip_atomic_fetch_add(p, v, __ATOMIC_RELAXED, __HIP_MEMORY_SCOPE_AGENT);
}

// ----------------------- degree / normalization -----------------------
__global__ void k_set1(float* __restrict__ p, int n) {
  int i = blockIdx.x * blockDim.x + threadIdx.x;
  if (i < n) p[i] = 1.0f;   // self-loop contributes 1 to degree
}
__global__ void k_count(const int* __restrict__ dst, float* __restrict__ deg, int e) {
  int i = blockIdx.x * blockDim.x + threadIdx.x;
  if (i < e) atomAddF(&deg[dst[i]], 1.0f);
}
__global__ void k_rsqrt(float* __restrict__ p, int n) {
  int i = blockIdx.x * blockDim.x + threadIdx.x;
  if (i < n) p[i] = rsqrtf(p[i]);   // dis = deg^-1/2 (in place)
}

// ----------------------- fp32 WMMA GEMM: H = X @ W --------------------
// One wave per 16x16 output tile; K stepped by 4 with V_WMMA_F32_16X16X4_F32.
// A 16x4 layout : lane l -> row = l%16 ; VGPR j -> K = k0 + 2*(l/16) + j
// B 4x16 layout : lane l -> col = l%16 ; VGPR j -> K = k0 + 2*(l/16) + j
// C/D 16x16     : lane l, VGPR v -> m = v + 8*(l/16), n = l%16
__global__ void k_gemm_wmma(const float* __restrict__ X, const float* __restrict__ W,
                            float* __restrict__ H, int n, int k, int m) {
  int wid  = (blockIdx.x * blockDim.x + threadIdx.x) >> 5;
  int lane = threadIdx.x & 31;
  int tilesN = m >> 4;
  int ntiles = (n >> 4) * tilesN;
  if (wid >= ntiles) return;                 // whole-wave uniform exit, EXEC stays all-1
  int tm = wid / tilesN;
  int tn = wid - tm * tilesN;
  int half = lane >> 4;
  int lm   = lane & 15;

  const float* xr = X + (size_t)(tm * 16 + lm) * (size_t)k;  // A row for this lane
  const float* wc = W + tn * 16 + lm;                        // B column for this lane

  v8f acc = {0.f, 0.f, 0.f, 0.f, 0.f, 0.f, 0.f, 0.f};
  for (int k0 = 0; k0 < k; k0 += 4) {
    int ka = k0 + 2 * half;
    v2f a; a.x = xr[ka];               a.y = xr[ka + 1];
    v2f b; b.x = wc[(size_t)ka * m];   b.y = wc[(size_t)(ka + 1) * m];
    acc = __builtin_amdgcn_wmma_f32_16x16x4_f32(false, a, false, b,
                                                (short)0, acc, false, false);
  }

  int m0 = tm * 16 + 8 * half;
  float* out = H + tn * 16 + lm;
#pragma unroll
  for (int v = 0; v < 8; ++v) out[(size_t)(m0 + v) * m] = acc[v];
}

// ----------------------- edge scatter (F=128) --------------------------
// One wave per edge; each lane owns 4 consecutive features (float4 of h row).
__global__ void k_scatter128(const int* __restrict__ src, const int* __restrict__ dst,
                             const float* __restrict__ h, const float* __restrict__ dis,
                             float* __restrict__ agg, int e) {
  int wid  = (blockIdx.x * blockDim.x + threadIdx.x) >> 5;
  int lane = threadIdx.x & 31;
  if (wid >= e) return;
  int s = src[wid], d = dst[wid];
  float norm = dis[s] * dis[d];
  float4 v = ((const float4*)(h + (size_t)s * FHID))[lane];
  float* a = agg + (size_t)d * FHID + lane * 4;
  atomAddF(a + 0, v.x * norm);
  atomAddF(a + 1, v.y * norm);
  atomAddF(a + 2, v.z * norm);
  atomAddF(a + 3, v.w * norm);
}

// ----------------------- edge scatter (F=32) ---------------------------
__global__ void k_scatter32(const int* __restrict__ src, const int* __restrict__ dst,
                            const float* __restrict__ h, const float* __restrict__ dis,
                            float* __restrict__ agg, int e) {
  int wid  = (blockIdx.x * blockDim.x + threadIdx.x) >> 5;
  int lane = threadIdx.x & 31;
  if (wid >= e) return;
  int s = src[wid], d = dst[wid];
  float norm = dis[s] * dis[d];
  float v = h[(size_t)s * FOUT + lane];
  atomAddF(&agg[(size_t)d * FOUT + lane], v * norm);
}

// ----------------- z = agg + h*dis^2 + bias ; accumulate BN stats ------
__global__ void k_post_stats(float* __restrict__ z, const float* __restrict__ h,
                             const float* __restrict__ dis, const float* __restrict__ bias,
                             float* __restrict__ sums, int n) {
  long long total  = (long long)n * FHID;
  long long stride = (long long)gridDim.x * blockDim.x;     // multiple of 128
  long long idx    = (long long)blockIdx.x * blockDim.x + threadIdx.x;
  int f = (int)(idx & (FHID - 1));                          // fixed feature per thread
  float bf = bias[f];
  float s1 = 0.f, s2 = 0.f;
  for (long long i = idx; i < total; i += stride) {
    int row = (int)(i >> 7);
    float dr = dis[row];
    float val = z[i] + h[i] * dr * dr + bf;
    z[i] = val;
    s1 += val;
    s2 += val * val;
  }
  atomAddF(&sums[f], s1);
  atomAddF(&sums[FHID + f], s2);
}

// ----------------- fold BN into affine: a[f], c[f] ---------------------
__global__ void k_bn_coef(float* __restrict__ sums, const float* __restrict__ g,
                          const float* __restrict__ be, int n) {
  int f = threadIdx.x;
  if (f < FHID) {
    float inv_n = 1.0f / (float)n;
    float mu  = sums[f] * inv_n;
    float var = sums[FHID + f] * inv_n - mu * mu;   // biased variance (torch BN)
    float inv = rsqrtf(var + BN_EPS);
    float a = g[f] * inv;
    sums[f]        = a;
    sums[FHID + f] = be[f] - mu * a;
  }
}

// ----------------- xout = relu(z*a + c) + xin --------------------------
__global__ void k_bnrelu_res(const float* __restrict__ z, const float* __restrict__ coef,
                             const float* __restrict__ xin, float* __restrict__ xout,
                             long long total) {
  long long i = (long long)blockIdx.x * blockDim.x + threadIdx.x;
  if (i < total) {
    int f = (int)(i & (FHID - 1));
    float v = z[i] * coef[f] + coef[FHID + f];
    v = v > 0.f ? v : 0.f;
    xout[i] = v + xin[i];
  }
}

// ----------------- final: z = agg + h*dis^2 + b ; log_softmax ----------
// One wave per node: 32 lanes == 32 classes; shuffle reductions.
__global__ void k_final(const float* __restrict__ z, const float* __restrict__ h,
                        const float* __restrict__ dis, const float* __restrict__ b,
                        float* __restrict__ out, int n) {
  int wid  = (blockIdx.x * blockDim.x + threadIdx.x) >> 5;
  int lane = threadIdx.x & 31;
  if (wid >= n) return;
  float dr = dis[wid];
  long long base = (long long)wid * FOUT;
  float v = z[base + lane] + h[base + lane] * dr * dr + b[lane];
  float m = v;
#pragma unroll
  for (int o = 16; o > 0; o >>= 1) m = fmaxf(m, __shfl_xor(m, o, 32));
  float e = expf(v - m);
  float s = e;
#pragma unroll
  for (int o = 16; o > 0; o >>= 1) s += __shfl_xor(s, o, 32);
  out[base + lane] = v - m - logf(s);
}

// ======================================================================
extern "C" void kernel_launch(void* const* d_in, const int* in_sizes, int n_in,
                              void* d_out, int out_size, void* d_ws, size_t ws_size,
                              hipStream_t stream) {
  const float* x   = (const float*)d_in[0];
  const int*   ei  = (const int*)d_in[1];
  const float* W0  = (const float*)d_in[2];
  const float* b0  = (const float*)d_in[3];
  const float* g0  = (const float*)d_in[4];
  const float* be0 = (const float*)d_in[5];
  const float* W1  = (const float*)d_in[6];
  const float* b1  = (const float*)d_in[7];
  const float* g1  = (const float*)d_in[8];
  const float* be1 = (const float*)d_in[9];
  const float* W2  = (const float*)d_in[10];
  const float* b2  = (const float*)d_in[11];

  const int N = in_sizes[0] / FIN;
  const int E = in_sizes[1] / 2;
  const int* src = ei;
  const int* dst = ei + E;

  // workspace layout (floats): dis[N] | h[N*128] | agg[N*128] | xbuf[N*128] | sums[256]
  float* ws   = (float*)d_ws;
  float* dis  = ws;
  float* h    = dis + N;
  float* agg  = h   + (size_t)N * FHID;
  float* xbuf = agg + (size_t)N * FHID;
  float* sums = xbuf + (size_t)N * FHID;

  const int T = 256;                                   // 8 wave32s per block
  const long long tot128 = (long long)N * FHID;
  const int gemmGrid128 = ((N / 16) * (FHID / 16) + 7) / 8;
  const int gemmGrid32  = ((N / 16) * (FOUT / 16) + 7) / 8;
  const int scatGrid    = (E + 7) / 8;
  const int elemGrid    = (int)((tot128 + T - 1) / T);

  // shared degree normalization: dis = (in_degree + 1)^-1/2
  k_set1 <<<(N + T - 1) / T, T, 0, stream>>>(dis, N);
  k_count<<<(E + T - 1) / T, T, 0, stream>>>(dst, dis, E);
  k_rsqrt<<<(N + T - 1) / T, T, 0, stream>>>(dis, N);

  // ---------------- layer 0 ----------------
  k_gemm_wmma<<<gemmGrid128, T, 0, stream>>>(x, W0, h, N, FIN, FHID);
  hipMemsetAsync(agg, 0, (size_t)N * FHID * sizeof(float), stream);
  hipMemsetAsync(sums, 0, 2 * FHID * sizeof(float), stream);
  k_scatter128<<<scatGrid, T, 0, stream>>>(src, dst, h, dis, agg, E);
  k_post_stats<<<1024, T, 0, stream>>>(agg, h, dis, b0, sums, N);
  k_bn_coef<<<1, FHID, 0, stream>>>(sums, g0, be0, N);
  k_bnrelu_res<<<elemGrid, T, 0, stream>>>(agg, sums, x, xbuf, tot128);

  // ---------------- layer 1 ----------------
  k_gemm_wmma<<<gemmGrid128, T, 0, stream>>>(xbuf, W1, h, N, FHID, FHID);
  hipMemsetAsync(agg, 0, (size_t)N * FHID * sizeof(float), stream);
  hipMemsetAsync(sums, 0, 2 * FHID * sizeof(float), stream);
  k_scatter128<<<scatGrid, T, 0, stream>>>(src, dst, h, dis, agg, E);
  k_post_stats<<<1024, T, 0, stream>>>(agg, h, dis, b1, sums, N);
  k_bn_coef<<<1, FHID, 0, stream>>>(sums, g1, be1, N);
  k_bnrelu_res<<<elemGrid, T, 0, stream>>>(agg, sums, xbuf, xbuf, tot128);

  // ---------------- layer 2 + log_softmax ----------------
  k_gemm_wmma<<<gemmGrid32, T, 0, stream>>>(xbuf, W2, h, N, FHID, FOUT);
  hipMemsetAsync(agg, 0, (size_t)N * FOUT * sizeof(float), stream);
  k_scatter32<<<scatGrid, T, 0, stream>>>(src, dst, h, dis, agg, E);
  k_final<<<(N + 7) / 8, T, 0, stream>>>(agg, h, dis, b2, (float*)d_out, N);
}